// ATTEN_61237643706783
// MI455X (gfx1250) — compile-verified
//
#include <hip/hip_runtime.h>
#include <hip/hip_bf16.h>
#include <math.h>

typedef float v2f __attribute__((ext_vector_type(2)));
typedef float v8f __attribute__((ext_vector_type(8)));

#define NROWS  8192
#define NFEATC 512
#define NHID   64
#define NHEADS 8
#define NCLS   40

// ---------------------------------------------------------------------------
// Pack B[K,ncols] (per head) into Bp[h][K/2][NCP] float2 pairs
// Bp[e] = { B[2*kp][col], B[2*kp+1][col] },  e = (h*(K/2)+kp)*NCP + col,
// zero-padded for col >= ncols.  This makes every WMMA B-fragment load a
// single unconditional global_load_b64.
// ---------------------------------------------------------------------------
__global__ void packB_kernel(const float* __restrict__ B, float* __restrict__ Bp,
                             int K, int ncols, int NCP, int total, int bHeadStride)
{
    const int gid = blockIdx.x * 256 + threadIdx.x;
    if (gid >= total) return;
    const int col = gid % NCP;
    const int t   = gid / NCP;
    const int kp  = t % (K / 2);
    const int h   = t / (K / 2);
    const float* Bh = B + (size_t)h * bHeadStride;
    float b0 = 0.f, b1 = 0.f;
    if (col < ncols) {
        b0 = Bh[(size_t)(2 * kp) * ncols + col];
        b1 = Bh[(size_t)(2 * kp + 1) * ncols + col];
    }
    Bp[(size_t)gid * 2]     = b0;
    Bp[(size_t)gid * 2 + 1] = b1;
}

// ---------------------------------------------------------------------------
// WMMA GEMM: C[h] = A @ B[h].  A: [NROWS,K] f32 row-major, Bp: packed (above),
// C: [NROWS,NC].  One wave -> one 16-row tile x (NT*16) cols.
// f32 WMMA 16x16x4 fragments: A/B = v2f (lane<16: K0,K1 ; lane>=16: K2,K3),
// C/D: 8 VGPRs, row = v + 8*(lane>=16), col = lane&15 (+16*j).
// ---------------------------------------------------------------------------
template <int NT, int NC>
__global__ void gemm16_wmma(const float* __restrict__ A,
                            const float* __restrict__ Bp,
                            float* __restrict__ C,
                            int K, int cHeadStride)
{
    constexpr int NCP = NT * 16;
    const int lane  = threadIdx.x & 31;
    const int wave  = threadIdx.x >> 5;
    const int tileM = blockIdx.x * 4 + wave;         // 4 waves per block
    const int head  = blockIdx.z;
    const int half  = lane >> 4;
    const int l16   = lane & 15;
    const int rowA  = tileM * 16 + l16;

    const v2f* __restrict__ Ap = (const v2f*)(A + (size_t)rowA * K) + half;
    const v2f* __restrict__ Bq = (const v2f*)Bp +
        (size_t)head * (K / 2) * NCP + (size_t)half * NCP + l16;

    v8f c0, c1, c2, c3;
#pragma unroll
    for (int v = 0; v < 8; ++v) { c0[v] = 0.f; c1[v] = 0.f; c2[v] = 0.f; c3[v] = 0.f; }

    for (int k0 = 0; k0 < K; k0 += 4) {
        const v2f a  = *Ap;
        const v2f b0 = Bq[0];
        c0 = __builtin_amdgcn_wmma_f32_16x16x4_f32(false, a, false, b0, (short)0, c0, false, false);
        if (NT > 1) {
            const v2f b1 = Bq[16];
            c1 = __builtin_amdgcn_wmma_f32_16x16x4_f32(false, a, false, b1, (short)0, c1, false, false);
        }
        if (NT > 2) {
            const v2f b2 = Bq[32];
            c2 = __builtin_amdgcn_wmma_f32_16x16x4_f32(false, a, false, b2, (short)0, c2, false, false);
        }
        if (NT > 3) {
            const v2f b3 = Bq[48];
            c3 = __builtin_amdgcn_wmma_f32_16x16x4_f32(false, a, false, b3, (short)0, c3, false, false);
        }
        Ap += 2;
        Bq += 2 * NCP;
    }

    float* Ch = C + (size_t)head * cHeadStride;
    const int rbase = tileM * 16 + 8 * half;
    {
        const int col = l16;
#pragma unroll
        for (int v = 0; v < 8; ++v) Ch[(size_t)(rbase + v) * NC + col] = c0[v];
    }
    if (NT > 1) {
        const int col = 16 + l16;
#pragma unroll
        for (int v = 0; v < 8; ++v) Ch[(size_t)(rbase + v) * NC + col] = c1[v];
    }
    if (NT > 2) {
        const int col = 32 + l16;
        if (col < NC)
#pragma unroll
            for (int v = 0; v < 8; ++v) Ch[(size_t)(rbase + v) * NC + col] = c2[v];
    }
    if (NT > 3) {
        const int col = 48 + l16;
        if (col < NC)
#pragma unroll
            for (int v = 0; v < 8; ++v) Ch[(size_t)(rbase + v) * NC + col] = c3[v];
    }
}

// ---------------------------------------------------------------------------
// f1[h,i] = Wh[h,i,:] . a1 ; f2[h,i] = Wh[h,i,:] . a2
// ---------------------------------------------------------------------------
__global__ void f12_kernel(const float* __restrict__ Wh,
                           const float* __restrict__ avec,
                           float* __restrict__ f1, float* __restrict__ f2,
                           int F, int H, int whHeadStride, int aHeadStride)
{
    const int gid = blockIdx.x * blockDim.x + threadIdx.x;
    if (gid >= H * NROWS) return;
    const int h = gid >> 13;            // NROWS == 8192
    const int i = gid & (NROWS - 1);
    const float* row = Wh + (size_t)h * whHeadStride + (size_t)i * F;
    const float* a1  = avec + (size_t)h * aHeadStride;
    const float* a2  = a1 + F;
    float s1 = 0.f, s2 = 0.f;
    for (int f = 0; f < F; ++f) { const float v = row[f]; s1 += v * a1[f]; s2 += v * a2[f]; }
    f1[gid] = s1;
    f2[gid] = s2;
}

// ---------------------------------------------------------------------------
// Per-head column mean of Wh (all-masked softmax fallback = uniform average)
// ---------------------------------------------------------------------------
__global__ void colmean_kernel(const float* __restrict__ Wh, float* __restrict__ mean,
                               int F, int whHeadStride)
{
    __shared__ float red[256];
    const int h = blockIdx.x, tid = threadIdx.x;
    const int f = tid & 63, p = tid >> 6;       // 4 partial sums per column
    float s = 0.f;
    if (f < F)
        for (int i = p; i < NROWS; i += 4)
            s += Wh[(size_t)h * whHeadStride + (size_t)i * F + f];
    red[tid] = s;
    __syncthreads();
    if (p == 0 && f < F)
        mean[(size_t)h * F + f] =
            (red[f] + red[64 + f] + red[128 + f] + red[192 + f]) / (float)NROWS;
}

// ---------------------------------------------------------------------------
// Per-head max of f2 (softmax shift)
// ---------------------------------------------------------------------------
__global__ void maxf2_kernel(const float* __restrict__ f2, float* __restrict__ f2max)
{
    __shared__ float red[256];
    const int h = blockIdx.x, tid = threadIdx.x;
    float m = -INFINITY;
    for (int i = tid; i < NROWS; i += 256) m = fmaxf(m, f2[(size_t)h * NROWS + i]);
    red[tid] = m;
    __syncthreads();
    for (int s = 128; s > 0; s >>= 1) {
        if (tid < s) red[tid] = fmaxf(red[tid], red[tid + s]);
        __syncthreads();
    }
    if (tid == 0) f2max[h] = red[0];
}

// ---------------------------------------------------------------------------
// Per-head: bitonic sort of f2 (descending, with indices) fully in LDS,
// then w = exp(f2 - max) and inclusive scan Z.  One 1024-thread block / head.
// ---------------------------------------------------------------------------
__global__ __launch_bounds__(1024)
void sort_scan_kernel(const float* __restrict__ f2, const float* __restrict__ f2max,
                      float* __restrict__ f2sorted, int* __restrict__ perm,
                      float* __restrict__ wout, float* __restrict__ Zpre)
{
    __shared__ float vals[NROWS];
    __shared__ int   idx[NROWS];
    const int h = blockIdx.x, tid = threadIdx.x, nth = 1024;

    for (int i = tid; i < NROWS; i += nth) { vals[i] = f2[(size_t)h * NROWS + i]; idx[i] = i; }
    __syncthreads();

    for (int k = 2; k <= NROWS; k <<= 1) {
        for (int j = k >> 1; j > 0; j >>= 1) {
            for (int i = tid; i < NROWS; i += nth) {
                const int ixj = i ^ j;
                if (ixj > i) {
                    const bool desc = ((i & k) == 0);
                    const float vi = vals[i], vj = vals[ixj];
                    if (desc ? (vi < vj) : (vi > vj)) {
                        vals[i] = vj; vals[ixj] = vi;
                        const int t = idx[i]; idx[i] = idx[ixj]; idx[ixj] = t;
                    }
                }
            }
            __syncthreads();
        }
    }

    const float m = f2max[h];
    for (int i = tid; i < NROWS; i += nth) {
        f2sorted[(size_t)h * NROWS + i] = vals[i];
        perm[(size_t)h * NROWS + i]     = idx[i];
        const float w = __expf(vals[i] - m);
        vals[i] = w;
        wout[(size_t)h * NROWS + i] = w;
    }
    __syncthreads();

    // chunked inclusive scan of vals[0..8191]; reuse idx LDS as float scratch
    float* chunk = (float*)idx;
    const int CH = NROWS / nth;                 // 8 contiguous elements / thread
    float loc[8], s = 0.f;
#pragma unroll
    for (int c = 0; c < CH; ++c) { s += vals[tid * CH + c]; loc[c] = s; }
    chunk[tid] = s;
    __syncthreads();
    for (int off = 1; off < nth; off <<= 1) {   // Hillis-Steele, 2-barrier
        const float t = (tid >= off) ? chunk[tid - off] : 0.f;
        __syncthreads();
        if (tid >= off) chunk[tid] += t;
        __syncthreads();
    }
    const float excl = (tid > 0) ? chunk[tid - 1] : 0.f;
#pragma unroll
    for (int c = 0; c < CH; ++c)
        Zpre[(size_t)h * NROWS + tid * CH + c] = excl + loc[c];
}

// ---------------------------------------------------------------------------
// Wp[h,k,f] = w[h,k] * Wh[h, perm[h,k], f]
// ---------------------------------------------------------------------------
__global__ void permw_kernel(const float* __restrict__ Wh, const float* __restrict__ w,
                             const int* __restrict__ perm, float* __restrict__ Wp,
                             int F, long long total, int whHeadStride)
{
    const long long gid = (long long)blockIdx.x * blockDim.x + threadIdx.x;
    if (gid >= total) return;
    const int f = (int)(gid % F);
    const long long r = gid / F;
    const int k = (int)(r & (NROWS - 1));
    const int h = (int)(r >> 13);
    const int j = perm[(size_t)h * NROWS + k];
    Wp[gid] = w[(size_t)h * NROWS + k] * Wh[(size_t)h * whHeadStride + (size_t)j * F + f];
}

// ---------------------------------------------------------------------------
// In-place inclusive prefix sum along k for each (h,f) column of Wp[h,:,F]
// One 256-thread block per column, 32 elements per thread kept in registers.
// ---------------------------------------------------------------------------
__global__ void colscan_kernel(float* __restrict__ Wp, int F)
{
    __shared__ float chunk[256];
    const int col = blockIdx.x;            // h*F + f
    const int h = col / F, f = col % F;
    float* base = Wp + (size_t)h * NROWS * F + f;
    const int tid = threadIdx.x;
    const int CH = NROWS / 256;            // 32
    float loc[32], s = 0.f;
#pragma unroll
    for (int c = 0; c < CH; ++c) { s += base[(size_t)(tid * CH + c) * F]; loc[c] = s; }
    chunk[tid] = s;
    __syncthreads();
    for (int off = 1; off < 256; off <<= 1) {
        const float t = (tid >= off) ? chunk[tid - off] : 0.f;
        __syncthreads();
        if (tid >= off) chunk[tid] += t;
        __syncthreads();
    }
    const float excl = (tid > 0) ? chunk[tid - 1] : 0.f;
#pragma unroll
    for (int c = 0; c < CH; ++c) base[(size_t)(tid * CH + c) * F] = excl + loc[c];
}

__device__ __forceinline__ int count_above(const float* __restrict__ d, float thr)
{
    // descending array: count of elements strictly > thr
    int lo = 0, hi = NROWS;
    while (lo < hi) {
        const int mid = (lo + hi) >> 1;
        if (d[mid] > thr) lo = mid + 1; else hi = mid;
    }
    return lo;
}

// ---------------------------------------------------------------------------
// Layer-1 finalize: h' = S/Z (or column mean), ELU, write concat h.
// ---------------------------------------------------------------------------
__global__ void finalize1_kernel(const float* __restrict__ f1,
                                 const float* __restrict__ f2sorted,
                                 const float* __restrict__ Zpre,
                                 const float* __restrict__ S,
                                 const float* __restrict__ meanWh,
                                 float* __restrict__ hcat)
{
    const long long gid = (long long)blockIdx.x * blockDim.x + threadIdx.x;
    if (gid >= (long long)NHEADS * NROWS * NHID) return;
    const int f = (int)(gid & 63);
    const long long r = gid >> 6;
    const int i = (int)(r & (NROWS - 1));
    const int h = (int)(r >> 13);
    const float thr = -f1[(size_t)h * NROWS + i];
    const int c = count_above(f2sorted + (size_t)h * NROWS, thr);
    float y;
    if (c == 0) y = meanWh[(size_t)h * NHID + f];
    else y = S[(size_t)h * NROWS * NHID + (size_t)(c - 1) * NHID + f] /
             Zpre[(size_t)h * NROWS + (c - 1)];
    y = (y > 0.f) ? y : expm1f(y);                  // ELU
    hcat[(size_t)i * (NHEADS * NHID) + (size_t)h * NHID + f] = y;
}

// ---------------------------------------------------------------------------
// Layer-2 finalize: h'' = S/Z (or mean), ELU, log_softmax over 40 classes.
// One 64-thread block per row.
// ---------------------------------------------------------------------------
__global__ void finalize2_kernel(const float* __restrict__ f1o,
                                 const float* __restrict__ f2sorted,
                                 const float* __restrict__ Zpre,
                                 const float* __restrict__ S,
                                 const float* __restrict__ meanWh,
                                 float* __restrict__ out)
{
    __shared__ float red[64];
    const int i = blockIdx.x, c = threadIdx.x;
    float yv = 0.f;
    if (c < NCLS) {
        const float thr = -f1o[i];
        const int cnt = count_above(f2sorted, thr);
        if (cnt == 0) yv = meanWh[c];
        else yv = S[(size_t)(cnt - 1) * NCLS + c] / Zpre[cnt - 1];
        yv = (yv > 0.f) ? yv : expm1f(yv);          // ELU
    }
    red[c] = (c < NCLS) ? yv : -INFINITY;
    __syncthreads();
    for (int s = 32; s > 0; s >>= 1) {
        if (c < s) red[c] = fmaxf(red[c], red[c + s]);
        __syncthreads();
    }
    const float m = red[0];
    __syncthreads();
    red[c] = (c < NCLS) ? __expf(yv - m) : 0.f;
    __syncthreads();
    for (int s = 32; s > 0; s >>= 1) {
        if (c < s) red[c] += red[c + s];
        __syncthreads();
    }
    const float lse = logf(red[0]);
    if (c < NCLS) out[(size_t)i * NCLS + c] = yv - m - lse;
}

// ---------------------------------------------------------------------------
extern "C" void kernel_launch(void* const* d_in, const int* in_sizes, int n_in,
                              void* d_out, int out_size, void* d_ws, size_t ws_size,
                              hipStream_t stream)
{
    (void)in_sizes; (void)n_in; (void)out_size; (void)ws_size;
    const float* x       = (const float*)d_in[0];
    /* adj (d_in[1]) is unused by the reference math */
    const float* W_heads = (const float*)d_in[2];
    const float* a_heads = (const float*)d_in[3];
    const float* W_out   = (const float*)d_in[4];
    const float* a_out   = (const float*)d_in[5];
    float* out = (float*)d_out;

    char* p = (char*)d_ws;
    auto alloc = [&](size_t bytes) -> void* {
        void* r = (void*)p;
        p += (bytes + 255) & ~(size_t)255;
        return r;
    };
    float* Wh1   = (float*)alloc((size_t)NHEADS * NROWS * NHID * 4);
    float* f1_1  = (float*)alloc((size_t)NHEADS * NROWS * 4);
    float* f2_1  = (float*)alloc((size_t)NHEADS * NROWS * 4);
    float* f2mx1 = (float*)alloc(NHEADS * 4);
    float* mean1 = (float*)alloc(NHEADS * NHID * 4);
    float* f2s1  = (float*)alloc((size_t)NHEADS * NROWS * 4);
    int*   perm1 = (int*)  alloc((size_t)NHEADS * NROWS * 4);
    float* w1    = (float*)alloc((size_t)NHEADS * NROWS * 4);
    float* Zpre1 = (float*)alloc((size_t)NHEADS * NROWS * 4);
    float* S1    = (float*)alloc((size_t)NHEADS * NROWS * NHID * 4);
    float* hcat  = (float*)alloc((size_t)NROWS * NFEATC * 4);
    float* Wh2   = (float*)alloc((size_t)NROWS * NCLS * 4);
    float* f1_2  = (float*)alloc(NROWS * 4);
    float* f2_2  = (float*)alloc(NROWS * 4);
    float* f2mx2 = (float*)alloc(4);
    float* mean2 = (float*)alloc(NCLS * 4);
    float* f2s2  = (float*)alloc(NROWS * 4);
    int*   perm2 = (int*)  alloc(NROWS * 4);
    float* w2    = (float*)alloc(NROWS * 4);
    float* Zpre2 = (float*)alloc(NROWS * 4);
    float* S2    = (float*)alloc((size_t)NROWS * NCLS * 4);
    float* Bp1   = (float*)alloc((size_t)NHEADS * (NFEATC / 2) * 64 * 2 * 4);
    float* Bp2   = (float*)alloc((size_t)(NFEATC / 2) * 48 * 2 * 4);

    // ---------------- Layer 1 (8 heads) ----------------
    {
        const int total = NHEADS * (NFEATC / 2) * 64;   // 131072
        packB_kernel<<<(total + 255) / 256, 256, 0, stream>>>(
            W_heads, Bp1, NFEATC, NHID, 64, total, NFEATC * NHID);
    }
    gemm16_wmma<4, NHID><<<dim3(NROWS / 64, 1, NHEADS), 128, 0, stream>>>(
        x, Bp1, Wh1, NFEATC, NROWS * NHID);
    f12_kernel<<<(NHEADS * NROWS) / 256, 256, 0, stream>>>(
        Wh1, a_heads, f1_1, f2_1, NHID, NHEADS, NROWS * NHID, 2 * NHID);
    maxf2_kernel<<<NHEADS, 256, 0, stream>>>(f2_1, f2mx1);
    colmean_kernel<<<NHEADS, 256, 0, stream>>>(Wh1, mean1, NHID, NROWS * NHID);
    sort_scan_kernel<<<NHEADS, 1024, 0, stream>>>(f2_1, f2mx1, f2s1, perm1, w1, Zpre1);
    {
        const long long total = (long long)NHEADS * NROWS * NHID;
        permw_kernel<<<(unsigned)((total + 255) / 256), 256, 0, stream>>>(
            Wh1, w1, perm1, S1, NHID, total, NROWS * NHID);
        colscan_kernel<<<NHEADS * NHID, 256, 0, stream>>>(S1, NHID);
        finalize1_kernel<<<(unsigned)((total + 255) / 256), 256, 0, stream>>>(
            f1_1, f2s1, Zpre1, S1, mean1, hcat);
    }

    // ---------------- Layer 2 (output) ----------------
    {
        const int total = (NFEATC / 2) * 48;            // 12288
        packB_kernel<<<(total + 255) / 256, 256, 0, stream>>>(
            W_out, Bp2, NFEATC, NCLS, 48, total, 0);
    }
    gemm16_wmma<3, NCLS><<<dim3(NROWS / 64, 1, 1), 128, 0, stream>>>(
        hcat, Bp2, Wh2, NFEATC, 0);
    f12_kernel<<<NROWS / 256, 256, 0, stream>>>(
        Wh2, a_out, f1_2, f2_2, NCLS, 1, NROWS * NCLS, 2 * NCLS);
    maxf2_kernel<<<1, 256, 0, stream>>>(f2_2, f2mx2);
    colmean_kernel<<<1, 256, 0, stream>>>(Wh2, mean2, NCLS, NROWS * NCLS);
    sort_scan_kernel<<<1, 1024, 0, stream>>>(f2_2, f2mx2, f2s2, perm2, w2, Zpre2);
    {
        const long long total = (long long)NROWS * NCLS;
        permw_kernel<<<(unsigned)((total + 255) / 256), 256, 0, stream>>>(
            Wh2, w2, perm2, S2, NCLS, total, NROWS * NCLS);
        colscan_kernel<<<NCLS, 256, 0, stream>>>(S2, NCLS);
    }
    finalize2_kernel<<<NROWS, 64, 0, stream>>>(f1_2, f2s2, Zpre2, S2, mean2, out);
}